// AnswerFilteringModule_57715770523975
// MI455X (gfx1250) — compile-verified
//
#include <hip/hip_runtime.h>
#include <hip/hip_bf16.h>

// ComplEx answer filtering on MI455X (gfx1250, wave32).
// scores = E(200000x512,f32) @ s(512,f32);  out = E[argmax(scores)].
// Bandwidth bound: 409.6 MB stream / 23.3 TB/s ~= 17.6 us floor.
// GEMV mapped onto V_WMMA_F32_16X16X4_F32 (keeps full f32 precision):
//   A = 16 entity rows x 4 K   (lane l: float2 E[row=l&15][k0+2*(l>>4)])
//   B = s broadcast across N   (lane l: float2 s[k0+2*(l>>4)] from LDS)
//   D column-invariant -> lanes (l&15)==0 carry the 16 scores per wave.

typedef __attribute__((ext_vector_type(2))) float v2f;
typedef __attribute__((ext_vector_type(8))) float v8f;

#define DIM 512
#define HALFD 256
#define NENT 200000
#define ROWS_PER_BLOCK 128   // 8 waves x 16 rows

// ---------------- kernel 1: build s vector, reset argmax slot ----------------
__global__ __launch_bounds__(256) void complex_prep_kernel(
    const float* __restrict__ H, const float* __restrict__ Q,
    float* __restrict__ svec, unsigned long long* __restrict__ best)
{
    const int i = threadIdx.x;                 // 0..255
    const float hre = H[i], him = H[i + HALFD];
    const float qre = Q[i], qim = Q[i + HALFD];
    svec[i]         = hre * qre - him * qim;   // s_re
    svec[i + HALFD] = hre * qim + him * qre;   // s_im
    if (i == 0) *best = 0ull;                  // keys are always > 0
}

// ---------------- kernel 2: WMMA GEMV + hierarchical argmax ----------------
__global__ __launch_bounds__(256) void complex_gemv_argmax_kernel(
    const float* __restrict__ E, const float* __restrict__ svec,
    unsigned long long* __restrict__ best)
{
    __shared__ float s_sh[DIM];
    __shared__ float sc[ROWS_PER_BLOCK];

    const int tid = threadIdx.x;
    s_sh[tid]         = svec[tid];
    s_sh[tid + HALFD] = svec[tid + HALFD];
    if (tid < ROWS_PER_BLOCK) sc[tid] = -__builtin_inff();
    __syncthreads();

    const int wave = tid >> 5;
    const int lane = tid & 31;
    const int half = lane >> 4;     // which 16-lane half
    const int m    = lane & 15;     // row within 16-row tile
    const int tile = blockIdx.x * 8 + wave;

    if (tile * 16 < NENT) {         // wave-uniform guard (EXEC stays all-ones)
        const int row = tile * 16 + m;
        const float* aptr = E + (size_t)row * DIM + 2 * half;
        const float* bptr = s_sh + 2 * half;

        v8f acc = {};
#pragma unroll 8
        for (int k0 = 0; k0 < DIM; k0 += 4) {
            // A tile: 8B per lane, lanes l / l+16 cover adjacent 8B of a row.
            v2f a = __builtin_nontemporal_load((const v2f*)aptr);
            // B tile: broadcast ds_load_b64 (two distinct addrs per wave).
            v2f b = *(const v2f*)bptr;
            acc = __builtin_amdgcn_wmma_f32_16x16x4_f32(
                false, a, false, b, (short)0, acc, false, false);
            aptr += 4;
            bptr += 4;
        }

        // Extract scores: VGPR v holds row v (lanes 0-15) / v+8 (lanes 16-31).
        if (m == 0) {
#pragma unroll
            for (int v = 0; v < 8; ++v)
                sc[wave * 16 + half * 8 + v] = acc[v];
        }
    }
    __syncthreads();

    if (tid == 0) {
        float bv = -__builtin_inff();
        int bj = 0;
        for (int j = 0; j < ROWS_PER_BLOCK; ++j) {
            float x = sc[j];
            if (x > bv) { bv = x; bj = j; }
        }
        unsigned int row = (unsigned int)blockIdx.x * ROWS_PER_BLOCK + (unsigned int)bj;
        if (row < NENT) {
            // Monotonic float->uint key; inverted index so max => lowest idx on tie.
            unsigned int u = __float_as_uint(bv);
            unsigned int key = (u & 0x80000000u) ? ~u : (u | 0x80000000u);
            unsigned long long packed =
                ((unsigned long long)key << 32) |
                (unsigned long long)(0xFFFFFFFFu - row);
            atomicMax(best, packed);
        }
    }
}

// ---------------- kernel 3: gather winning row ----------------
__global__ __launch_bounds__(256) void complex_gather_kernel(
    const float* __restrict__ E, const unsigned long long* __restrict__ best,
    float* __restrict__ out)
{
    const unsigned long long p = *best;
    const unsigned int row = 0xFFFFFFFFu - (unsigned int)(p & 0xFFFFFFFFull);
    const int i = threadIdx.x;
    const float* src = E + (size_t)row * DIM;
    out[i]         = src[i];
    out[i + HALFD] = src[i + HALFD];
}

extern "C" void kernel_launch(void* const* d_in, const int* in_sizes, int n_in,
                              void* d_out, int out_size, void* d_ws, size_t ws_size,
                              hipStream_t stream)
{
    const float* H = (const float*)d_in[0];   // head_entity      [512]
    const float* Q = (const float*)d_in[1];   // question_embed   [512]
    const float* E = (const float*)d_in[2];   // entity_embeds    [200000*512]
    float* out = (float*)d_out;               // [512]

    // workspace: [0,2048) s vector, [2048,2056) packed argmax slot
    float* svec = (float*)d_ws;
    unsigned long long* best =
        (unsigned long long*)((char*)d_ws + DIM * sizeof(float));

    complex_prep_kernel<<<1, 256, 0, stream>>>(H, Q, svec, best);

    const int nblocks = (NENT + ROWS_PER_BLOCK - 1) / ROWS_PER_BLOCK; // 1563
    complex_gemv_argmax_kernel<<<nblocks, 256, 0, stream>>>(E, svec, best);

    complex_gather_kernel<<<1, 256, 0, stream>>>(E, best, out);
}